// crossCorrelation3D_22952305230244
// MI455X (gfx1250) — compile-verified
//
#include <hip/hip_runtime.h>

// ---------------------------------------------------------------------------
// LNCC (local cross-correlation, 9x9x9 box window) for (2,1,160,192,160) f32.
// Three separable passes; the 9-tap box filter along each axis is computed as
// D = W * X with W a 16x24 banded ones matrix via V_WMMA_F32_16X16X4_F32
// (6 accumulating WMMAs per 16x16 output tile, one accumulator per field).
//
// Zero-padding is handled branch-free: out-of-range filter taps are loaded
// from a clamped (valid) address and the corresponding COLUMN of the banded
// W matrix is zeroed, so the garbage rows of B are multiplied by 0.0 inside
// the WMMA. This keeps all loads unconditional and batchable.
//
// Intermediate fields are stored as float4{I,T,II,TT} + float{IT} per voxel
// so passes 2/3 fetch each tap with one b128 + one b32 load.
// ---------------------------------------------------------------------------

typedef __attribute__((ext_vector_type(2))) float v2f;
typedef __attribute__((ext_vector_type(8))) float v8f;

#define DX 160
#define DY 192
#define DZ 160
#define NB 2
#define SY DX                 // y stride (elements)
#define SZ (DX * DY)          // z stride
#define NVOL (DX * DY * DZ)   // per-batch voxels
#define NTOT (NB * NVOL)      // 9,830,400
#define KSIZE_F 729.0f

#define WPB 8                 // waves per block (256 threads, wave32)
#define NTILES 38400          // tiles per pass
#define NBLK (NTILES / WPB)   // 4800 blocks per pass

// WMMA f32 16x16x4 accumulate: D += W_chunk(16x4) * B_chunk(4x16)
#define WMMA_ACC(acc, a, b) \
  acc = __builtin_amdgcn_wmma_f32_16x16x4_f32(false, (a), false, (b), (short)0, (acc), false, false)

__device__ __forceinline__ int iclamp(int v, int lo, int hi) {
  return v < lo ? lo : (v > hi ? hi : v);
}

// Banded ones filter matrix column, with the zero-pad mask folded in:
// W[m][k] = 1 iff (m <= k <= m+8) AND tap position k is inside the volume.
__device__ __forceinline__ float wband(int m, int k, bool inb) {
  return (k >= m && k <= m + 8 && inb) ? 1.0f : 0.0f;
}

// ---------------------------------------------------------------------------
// Pass 1: pointwise fields + box-sum along X (stride-1 axis).
// Tile: 16 x-outputs (filter axis) x 16 y-columns at fixed (b, z).
// ---------------------------------------------------------------------------
__global__ __launch_bounds__(256) void lncc_pass_x(
    const float* __restrict__ in, const float* __restrict__ tg,
    float4* __restrict__ A4, float* __restrict__ A1) {
  int wg   = blockIdx.x * WPB + (threadIdx.x >> 5);
  int lane = threadIdx.x & 31;
  int half = lane >> 4;   // 0: lanes 0-15, 1: lanes 16-31
  int n    = lane & 15;   // B/D column index (and A row index m)

  int t  = wg;
  int xs = t % 10;  t /= 10;   // x segment (160/16)
  int ys = t % 12;  t /= 12;   // y tile    (192/16)
  int z  = t % DZ;  t /= DZ;
  int b  = t;

  int colOff = b * NVOL + z * SZ + (ys * 16 + n) * SY;  // fixed y column
  int aStart = xs * 16 - 4;                             // first needed x (halo)

  v8f acc0 = {}, acc1 = {}, acc2 = {}, acc3 = {}, acc4 = {};

#pragma unroll
  for (int c = 0; c < 6; ++c) {
    int k0 = 4 * c + 2 * half;      // K slot held by this half-wave, VGPR0
    int k1 = k0 + 1;                // VGPR1
    int p0 = aStart + k0;           // x position of tap k0 (k1 = p0+1)
    bool ok0 = (unsigned)p0 < (unsigned)DX;
    bool ok1 = (unsigned)(p0 + 1) < (unsigned)DX;
    v2f a;
    a.x = wband(n, k0, ok0);
    a.y = wband(n, k1, ok1);
    // Clamped, unconditional, 8-byte-aligned pair loads (p0 is even).
    int pc = iclamp(p0, 0, DX - 2);
    float2 iv = *(const float2*)(in + colOff + pc);
    float2 gv = *(const float2*)(tg + colOff + pc);
    float i0 = iv.x, i1 = iv.y;
    float t0 = (gv.x + 1.0f) * 0.5f;   // target rescale (-1,1)->(0,1)
    float t1 = (gv.y + 1.0f) * 0.5f;
    v2f bI  = {i0, i1};
    v2f bT  = {t0, t1};
    v2f bII = {i0 * i0, i1 * i1};
    v2f bTT = {t0 * t0, t1 * t1};
    v2f bIT = {i0 * t0, i1 * t1};
    WMMA_ACC(acc0, a, bI);
    WMMA_ACC(acc1, a, bT);
    WMMA_ACC(acc2, a, bII);
    WMMA_ACC(acc3, a, bTT);
    WMMA_ACC(acc4, a, bIT);
  }

  // D layout: VGPR r -> x = xs*16 + r + 8*half, column y fixed per lane.
  int aOut = xs * 16 + 8 * half;
#pragma unroll
  for (int r = 0; r < 8; ++r) {
    int o = colOff + aOut + r;
    A4[o] = make_float4(acc0[r], acc1[r], acc2[r], acc3[r]);
    A1[o] = acc4[r];
  }
}

// ---------------------------------------------------------------------------
// Pass 2: box-sum along Y on the 5 fields.
// Tile: 16 y-outputs (filter axis) x 16 x-columns at fixed (b, z).
// ---------------------------------------------------------------------------
__global__ __launch_bounds__(256) void lncc_pass_y(
    const float4* __restrict__ A4, const float* __restrict__ A1,
    float4* __restrict__ B4, float* __restrict__ B1) {
  int wg   = blockIdx.x * WPB + (threadIdx.x >> 5);
  int lane = threadIdx.x & 31;
  int half = lane >> 4;
  int n    = lane & 15;

  int t  = wg;
  int xs = t % 10;  t /= 10;
  int ys = t % 12;  t /= 12;
  int z  = t % DZ;  t /= DZ;
  int b  = t;

  int colOff = b * NVOL + z * SZ + xs * 16 + n;  // fixed x column (stride 1)
  int aStart = ys * 16 - 4;

  v8f acc0 = {}, acc1 = {}, acc2 = {}, acc3 = {}, acc4 = {};

#pragma unroll
  for (int c = 0; c < 6; ++c) {
    int k0 = 4 * c + 2 * half;
    int k1 = k0 + 1;
    int p0 = aStart + k0, p1 = aStart + k1;
    bool ok0 = (unsigned)p0 < (unsigned)DY;
    bool ok1 = (unsigned)p1 < (unsigned)DY;
    v2f a;
    a.x = wband(n, k0, ok0);
    a.y = wband(n, k1, ok1);
    int o0 = colOff + iclamp(p0, 0, DY - 1) * SY;
    int o1 = colOff + iclamp(p1, 0, DY - 1) * SY;
    float4 f0 = A4[o0];
    float4 f1 = A4[o1];
    float  e0 = A1[o0];
    float  e1 = A1[o1];
    v2f b0 = {f0.x, f1.x};
    v2f b1 = {f0.y, f1.y};
    v2f b2 = {f0.z, f1.z};
    v2f b3 = {f0.w, f1.w};
    v2f b4 = {e0, e1};
    WMMA_ACC(acc0, a, b0);
    WMMA_ACC(acc1, a, b1);
    WMMA_ACC(acc2, a, b2);
    WMMA_ACC(acc3, a, b3);
    WMMA_ACC(acc4, a, b4);
  }

  int aOut = ys * 16 + 8 * half;
#pragma unroll
  for (int r = 0; r < 8; ++r) {
    int o = colOff + (aOut + r) * SY;
    B4[o] = make_float4(acc0[r], acc1[r], acc2[r], acc3[r]);
    B1[o] = acc4[r];
  }
}

// ---------------------------------------------------------------------------
// Pass 3: box-sum along Z, fused cc math + deterministic block reduction.
// Tile: 16 z-outputs (filter axis) x 16 x-columns at fixed (b, y).
// ---------------------------------------------------------------------------
__global__ __launch_bounds__(256) void lncc_pass_z(
    const float4* __restrict__ B4, const float* __restrict__ B1,
    float* __restrict__ partial) {
  __shared__ float sred[WPB];

  int wave = threadIdx.x >> 5;
  int wg   = blockIdx.x * WPB + wave;
  int lane = threadIdx.x & 31;
  int half = lane >> 4;
  int n    = lane & 15;

  int t  = wg;
  int xs = t % 10;   t /= 10;    // x tile
  int zs = t % 10;   t /= 10;    // z segment (160/16)
  int y  = t % DY;   t /= DY;
  int b  = t;

  int colOff = b * NVOL + y * SY + xs * 16 + n;  // fixed (y, x) column
  int aStart = zs * 16 - 4;

  v8f acc0 = {}, acc1 = {}, acc2 = {}, acc3 = {}, acc4 = {};

#pragma unroll
  for (int c = 0; c < 6; ++c) {
    int k0 = 4 * c + 2 * half;
    int k1 = k0 + 1;
    int p0 = aStart + k0, p1 = aStart + k1;
    bool ok0 = (unsigned)p0 < (unsigned)DZ;
    bool ok1 = (unsigned)p1 < (unsigned)DZ;
    v2f a;
    a.x = wband(n, k0, ok0);
    a.y = wband(n, k1, ok1);
    int o0 = colOff + iclamp(p0, 0, DZ - 1) * SZ;
    int o1 = colOff + iclamp(p1, 0, DZ - 1) * SZ;
    float4 f0 = B4[o0];
    float4 f1 = B4[o1];
    float  e0 = B1[o0];
    float  e1 = B1[o1];
    v2f b0 = {f0.x, f1.x};
    v2f b1 = {f0.y, f1.y};
    v2f b2 = {f0.z, f1.z};
    v2f b3 = {f0.w, f1.w};
    v2f b4 = {e0, e1};
    WMMA_ACC(acc0, a, b0);
    WMMA_ACC(acc1, a, b1);
    WMMA_ACC(acc2, a, b2);
    WMMA_ACC(acc3, a, b3);
    WMMA_ACC(acc4, a, b4);
  }

  // cc math per output voxel (8 per lane), then wave/block reduction.
  float local = 0.0f;
#pragma unroll
  for (int r = 0; r < 8; ++r) {
    float I_s  = acc0[r];
    float T_s  = acc1[r];
    float II_s = acc2[r];
    float TT_s = acc3[r];
    float IT_s = acc4[r];
    float Ihat = I_s * (1.0f / KSIZE_F);
    float That = T_s * (1.0f / KSIZE_F);
    float cross = IT_s - Ihat * T_s - That * I_s + That * Ihat * KSIZE_F;
    float T_var = TT_s - 2.0f * That * T_s + That * That * KSIZE_F;
    float I_var = II_s - 2.0f * Ihat * I_s + Ihat * Ihat * KSIZE_F;
    local += cross * cross / (T_var * I_var + 1e-5f);
  }

  // wave32 butterfly reduction
#pragma unroll
  for (int off = 16; off > 0; off >>= 1)
    local += __shfl_xor(local, off, 32);

  if (lane == 0) sred[wave] = local;
  __syncthreads();
  if (threadIdx.x == 0) {
    float s = 0.0f;
#pragma unroll
    for (int i = 0; i < WPB; ++i) s += sred[i];  // fixed order -> deterministic
    partial[blockIdx.x] = s;
  }
}

// ---------------------------------------------------------------------------
// Final: fixed-order reduction of per-block partials -> scalar output.
// ---------------------------------------------------------------------------
__global__ __launch_bounds__(256) void lncc_final(const float* __restrict__ partial,
                                                  int np, float* __restrict__ out) {
  __shared__ float sdata[256];
  float s = 0.0f;
  for (int i = threadIdx.x; i < np; i += 256) s += partial[i];
  sdata[threadIdx.x] = s;
  __syncthreads();
  if (threadIdx.x == 0) {
    float tot = 0.0f;
    for (int i = 0; i < 256; ++i) tot += sdata[i];  // fixed order
    out[0] = -(tot / (float)NTOT);
  }
}

extern "C" void kernel_launch(void* const* d_in, const int* in_sizes, int n_in,
                              void* d_out, int out_size, void* d_ws, size_t ws_size,
                              hipStream_t stream) {
  (void)in_sizes; (void)n_in; (void)out_size; (void)ws_size;
  const float* in = (const float*)d_in[0];
  const float* tg = (const float*)d_in[1];
  float* ws = (float*)d_ws;

  // ws layout (floats): [4*NTOT] A4 | [NTOT] A1 | [4*NTOT] B4 | [NTOT] B1 | [NBLK] partials
  float4* A4 = (float4*)ws;
  float*  A1 = ws + (size_t)4 * NTOT;
  float4* B4 = (float4*)(ws + (size_t)5 * NTOT);
  float*  B1 = ws + (size_t)9 * NTOT;
  float*  P  = ws + (size_t)10 * NTOT;

  dim3 grid(NBLK), block(256);
  lncc_pass_x<<<grid, block, 0, stream>>>(in, tg, A4, A1);
  lncc_pass_y<<<grid, block, 0, stream>>>(A4, A1, B4, B1);
  lncc_pass_z<<<grid, block, 0, stream>>>(B4, B1, P);
  lncc_final<<<dim3(1), block, 0, stream>>>(P, NBLK, (float*)d_out);
}